// GetImportantStruct_51015621542220
// MI455X (gfx1250) — compile-verified
//
#include <hip/hip_runtime.h>
#include <math.h>

typedef __attribute__((ext_vector_type(2))) float v2f;
typedef __attribute__((ext_vector_type(8))) float v8f;

#define TPB 256
#define SCAN_T 256
#define SCAN_ITEMS 4
#define SCAN_TILE (SCAN_T * SCAN_ITEMS)

// ---------- small helpers ----------
__device__ __forceinline__ int find_seg(const int* __restrict__ off, int G, int e) {
  // largest g with off[g] <= e  (off = exclusive scan of edge_num)
  int lo = 0, hi = G - 1;
  while (lo < hi) {
    int mid = (lo + hi + 1) >> 1;
    if (off[mid] <= e) lo = mid; else hi = mid - 1;
  }
  return lo;
}

__global__ void fill_i32(int* __restrict__ p, int v, int n) {
  int i = blockIdx.x * blockDim.x + threadIdx.x;
  if (i < n) p[i] = v;
}

__global__ void hist_kernel(const int* __restrict__ lbl, int* __restrict__ cnt, int n) {
  int i = blockIdx.x * blockDim.x + threadIdx.x;
  if (i < n) atomicAdd(&cnt[lbl[i]], 1);
}

// ---------- exclusive scan: 3-kernel chain ----------
__global__ void scan_block_k(const int* __restrict__ in, int* __restrict__ out,
                             int* __restrict__ bsum, int n) {
  __shared__ int sh[SCAN_T];
  int t = threadIdx.x;
  int base = blockIdx.x * SCAN_TILE + t * SCAN_ITEMS;
  int pre[SCAN_ITEMS];
  int s = 0;
#pragma unroll
  for (int j = 0; j < SCAN_ITEMS; ++j) {
    int idx = base + j;
    int val = (idx < n) ? in[idx] : 0;
    pre[j] = s; s += val;
  }
  sh[t] = s;
  __syncthreads();
  for (int off = 1; off < SCAN_T; off <<= 1) {
    int xv = (t >= off) ? sh[t - off] : 0;
    __syncthreads();
    sh[t] += xv;
    __syncthreads();
  }
  int excl = (t > 0) ? sh[t - 1] : 0;
#pragma unroll
  for (int j = 0; j < SCAN_ITEMS; ++j) {
    int idx = base + j;
    if (idx < n) out[idx] = excl + pre[j];
  }
  if (t == SCAN_T - 1 && bsum) bsum[blockIdx.x] = sh[SCAN_T - 1];
}

__global__ void scan_top_k(int* __restrict__ bsum, int nb, int* __restrict__ total) {
  __shared__ int sh[SCAN_T];
  int t = threadIdx.x;
  int pre[SCAN_ITEMS];
  int s = 0;
#pragma unroll
  for (int j = 0; j < SCAN_ITEMS; ++j) {
    int idx = t * SCAN_ITEMS + j;
    int val = (idx < nb) ? bsum[idx] : 0;
    pre[j] = s; s += val;
  }
  sh[t] = s;
  __syncthreads();
  for (int off = 1; off < SCAN_T; off <<= 1) {
    int xv = (t >= off) ? sh[t - off] : 0;
    __syncthreads();
    sh[t] += xv;
    __syncthreads();
  }
  int excl = (t > 0) ? sh[t - 1] : 0;
#pragma unroll
  for (int j = 0; j < SCAN_ITEMS; ++j) {
    int idx = t * SCAN_ITEMS + j;
    if (idx < nb) bsum[idx] = excl + pre[j];
  }
  if (t == SCAN_T - 1 && total) *total = sh[SCAN_T - 1];
}

__global__ void scan_add_k(int* __restrict__ out, const int* __restrict__ bsum, int n) {
  int b = bsum[blockIdx.x];
  if (b == 0) return;
  int base = blockIdx.x * SCAN_TILE + threadIdx.x * SCAN_ITEMS;
#pragma unroll
  for (int j = 0; j < SCAN_ITEMS; ++j) {
    int idx = base + j;
    if (idx < n) out[idx] += b;
  }
}

// ---------- score GEMV via V_WMMA_F32_16X16X4_F32 ----------
// One wave32 per 16 nodes. A = 16x4 f32 tile of x (lanes 0-15: K=0,1 ; lanes
// 16-31: K=2,3). B = 4x16 with W replicated in every column, so every column
// of D equals the 16 dot products. Lanes 0/16 hold M=0..7 / M=8..15 in the 8
// accumulator VGPRs and write the scores.
__global__ __launch_bounds__(32) void score_wmma_k(const float* __restrict__ x,
                                                   const float* __restrict__ W,
                                                   float* __restrict__ score,
                                                   int N, int C) {
  const int lane = threadIdx.x;        // 0..31
  const int row  = lane & 15;
  const int half = lane >> 4;
  const int base = blockIdx.x * 16;
  int node = base + row;
  if (node >= N) node = N - 1;         // keep EXEC all-ones (dup rows are harmless)
  const float* __restrict__ xr = x + (size_t)node * C;

  v8f acc = {0.f, 0.f, 0.f, 0.f, 0.f, 0.f, 0.f, 0.f};
  const int C4 = C & ~3;
  for (int k = 0; k < C4; k += 4) {
    const int kk = k + 2 * half;
    v2f a; a.x = xr[kk]; a.y = xr[kk + 1];
    v2f b; b.x = W[kk];  b.y = W[kk + 1];
    acc = __builtin_amdgcn_wmma_f32_16x16x4_f32(false, a, false, b,
                                                (short)0, acc, false, false);
  }

  __shared__ float tail[16];
  if (C4 < C) {                         // generic tail (not hit for C=512)
    if (half == 0) {
      float t = 0.f;
      for (int k = C4; k < C; ++k) t += xr[k] * W[k];
      tail[row] = t;
    }
    __syncthreads();
  }

  if (row == 0) {                       // lanes 0 and 16
#pragma unroll
    for (int r = 0; r < 8; ++r) {
      const int i = base + half * 8 + r;
      if (i < N) {
        float d = acc[r];
        if (C4 < C) d += tail[half * 8 + r];
        const float m = d / (float)C;
        score[i] = 1.0f / (1.0f + expf(-m));
      }
    }
  }
}

// ---------- elementwise / graph kernels ----------
__global__ void lastnode_k(const int* __restrict__ counts, const int* __restrict__ node_off,
                           int* __restrict__ last_node, int* __restrict__ is_last,
                           int G, int N) {
  int g = blockIdx.x * blockDim.x + threadIdx.x;
  if (g >= G) return;
  int ln = node_off[g] + counts[g] - 1;
  last_node[g] = ln;
  if (ln >= 0 && ln < N) is_last[ln] = 1;
}

__global__ void low_k(const float* __restrict__ score, const int* __restrict__ is_last,
                      int* __restrict__ low, int N) {
  int i = blockIdx.x * blockDim.x + threadIdx.x;
  if (i < N) low[i] = (score[i] < 0.5f && !is_last[i]) ? 1 : 0;
}

__global__ void xout_vec4_k(const float4* __restrict__ x4, const float* __restrict__ score,
                            float4* __restrict__ out4, int n4, int c4) {
  int i = blockIdx.x * blockDim.x + threadIdx.x;
  if (i >= n4) return;
  const float s = score[i / c4] * 2.0f;
  float4 v = x4[i];
  v.x *= s; v.y *= s; v.z *= s; v.w *= s;
  out4[i] = v;
}

__global__ void xout_s_k(const float* __restrict__ x, const float* __restrict__ score,
                         float* __restrict__ out, long long total, int C) {
  long long i = (long long)blockIdx.x * blockDim.x + threadIdx.x;
  if (i >= total) return;
  out[i] = x[i] * score[i / C] * 2.0f;
}

__global__ void cand_k(const int* __restrict__ ei, const int* __restrict__ batch,
                       const int* __restrict__ e_off, const int* __restrict__ last_node,
                       const int* __restrict__ low, int* __restrict__ rem, int G, int E) {
  int e = blockIdx.x * blockDim.x + threadIdx.x;
  if (e >= E) return;
  const int g = find_seg(e_off, G, e);
  const int local = e - e_off[g];
  const int src = ei[e];
  const int dst = ei[E + e];
  if (dst == last_node[g] && low[src] && batch[src] == g && local >= 1)
    atomicMax(&rem[src], e);
}

__global__ void drop_k(const int* __restrict__ rem, int* __restrict__ emask, int N) {
  int i = blockIdx.x * blockDim.x + threadIdx.x;
  if (i >= N) return;
  int r = rem[i];
  if (r >= 0) emask[r] = 0;
}

__global__ void edge_idx_compact_k(const int* __restrict__ ei, const int* __restrict__ emask,
                                   const int* __restrict__ epos, const int* __restrict__ counters,
                                   float* __restrict__ dout, size_t baseNC, int E) {
  int e = blockIdx.x * blockDim.x + threadIdx.x;
  if (e >= E || !emask[e]) return;
  const int Ep = counters[0];
  const int p = epos[e];
  dout[baseNC + (size_t)p] = (float)ei[e];
  dout[baseNC + (size_t)Ep + p] = (float)ei[E + e];
}

__global__ void inform_compact_k(const float* __restrict__ inf_, const int* __restrict__ emask,
                                 const int* __restrict__ epos, const int* __restrict__ counters,
                                 float* __restrict__ dout, size_t baseNC, int E, int DE) {
  long long i = (long long)blockIdx.x * blockDim.x + threadIdx.x;
  const long long tot = (long long)E * DE;
  if (i >= tot) return;
  const int e = (int)(i / DE);
  const int j = (int)(i % DE);
  if (!emask[e]) return;
  const int Ep = counters[0];
  const size_t off = baseNC + 2 * (size_t)Ep + (size_t)epos[e] * DE + j;
  dout[off] = inf_[i];
}

__global__ void write_enp_k(const int* __restrict__ e_off, const int* __restrict__ edge_num,
                            const int* __restrict__ epos, const int* __restrict__ counters,
                            float* __restrict__ dout, size_t baseNC, int G, int E, int DE) {
  int g = blockIdx.x * blockDim.x + threadIdx.x;
  if (g >= G) return;
  const int Ep = counters[0];
  const int a = e_off[g];
  const int b = a + edge_num[g];
  const int sa = (a < E) ? epos[a] : Ep;
  const int sb = (b < E) ? epos[b] : Ep;
  const size_t off = baseNC + 2 * (size_t)Ep + (size_t)Ep * DE + g;
  dout[off] = (float)(sb - sa);
}

__global__ void brdrop_k(const int* __restrict__ low, const int* __restrict__ batch,
                         const int* __restrict__ node_off, const int* __restrict__ node_off2,
                         const int* __restrict__ br_com_num, int* __restrict__ br_mask,
                         int N, int N2) {
  int i = blockIdx.x * blockDim.x + threadIdx.x;
  if (i >= N || !low[i]) return;
  const int g = batch[i];
  const long long pos = (long long)i - node_off[g] + br_com_num[g] + node_off2[g];
  if (pos >= 0 && pos < N2) br_mask[pos] = 0;
}

__global__ void remap_k(const int* __restrict__ br_mask, const int* __restrict__ npos,
                        int* __restrict__ remap, int N2) {
  int i = blockIdx.x * blockDim.x + threadIdx.x;
  if (i < N2) remap[i] = br_mask[i] ? npos[i] : -1;
}

__global__ void brfeat_compact_k(const float* __restrict__ brf, const int* __restrict__ br_mask,
                                 const int* __restrict__ npos, const int* __restrict__ counters,
                                 float* __restrict__ dout, size_t baseNC,
                                 int G, int DE, int C, int N2) {
  long long idx = (long long)blockIdx.x * blockDim.x + threadIdx.x;
  const long long tot = (long long)N2 * C;
  if (idx >= tot) return;
  const int i = (int)(idx / C);
  const int j = (int)(idx % C);
  if (!br_mask[i]) return;
  const int Ep = counters[0];
  const size_t off4 = baseNC + 2 * (size_t)Ep + (size_t)Ep * DE + G;
  dout[off4 + (size_t)npos[i] * C + j] = brf[idx];
}

__global__ void keep_k(const int* __restrict__ bri, const int* __restrict__ remap,
                       int* __restrict__ kmask, int E2) {
  int e = blockIdx.x * blockDim.x + threadIdx.x;
  if (e >= E2) return;
  const int r = remap[bri[e]];
  const int c = remap[bri[E2 + e]];
  kmask[e] = (r >= 0 && c >= 0) ? 1 : 0;
}

__global__ void brindex_k(const int* __restrict__ bri, const int* __restrict__ remap,
                          const int* __restrict__ kmask, const int* __restrict__ kpos,
                          const int* __restrict__ counters, float* __restrict__ dout,
                          size_t baseNC, int G, int DE, int C, int E2) {
  int e = blockIdx.x * blockDim.x + threadIdx.x;
  if (e >= E2 || !kmask[e]) return;
  const int Ep = counters[0], Np = counters[1], Epp = counters[2];
  const size_t off5 = baseNC + 2 * (size_t)Ep + (size_t)Ep * DE + G + (size_t)Np * C;
  const int p = kpos[e];
  dout[off5 + (size_t)p] = (float)remap[bri[e]];
  dout[off5 + (size_t)Epp + p] = (float)remap[bri[E2 + e]];
}

__global__ void brbatch_k(const int* __restrict__ brb, const int* __restrict__ br_mask,
                          const int* __restrict__ npos, const int* __restrict__ counters,
                          float* __restrict__ dout, size_t baseNC, int G, int DE, int C, int N2) {
  int i = blockIdx.x * blockDim.x + threadIdx.x;
  if (i >= N2 || !br_mask[i]) return;
  const int Ep = counters[0], Np = counters[1], Epp = counters[2];
  const size_t off = baseNC + 2 * (size_t)Ep + (size_t)Ep * DE + G +
                     (size_t)Np * C + 2 * (size_t)Epp + npos[i];
  dout[off] = (float)brb[i];
}

// ---------- host ----------
static inline int cdiv_ll(long long a, long long b) { return (int)((a + b - 1) / b); }

extern "C" void kernel_launch(void* const* d_in, const int* in_sizes, int n_in,
                              void* d_out, int out_size, void* d_ws, size_t ws_size,
                              hipStream_t stream) {
  const float* x        = (const float*)d_in[0];
  const int*   ei       = (const int*)d_in[1];
  const int*   edge_num = (const int*)d_in[2];
  const float* einf     = (const float*)d_in[3];
  const int*   batch    = (const int*)d_in[4];
  const float* brf      = (const float*)d_in[5];
  const int*   bri      = (const int*)d_in[6];
  const int*   brb      = (const int*)d_in[7];
  const int*   brc      = (const int*)d_in[8];
  const float* W        = (const float*)d_in[9];

  const int G  = in_sizes[2];
  const int N  = in_sizes[4];
  const int C  = in_sizes[0] / N;
  const int E  = in_sizes[1] / 2;
  const int DE = in_sizes[3] / E;
  const int N2 = in_sizes[7];
  const int E2 = in_sizes[6] / 2;
  const size_t baseNC = (size_t)N * C;

  float* dout = (float*)d_out;

  // workspace layout (4-byte units)
  int* w = (int*)d_ws;
  float* score   = (float*)w; w += N;
  int* counts    = w; w += G;
  int* node_off  = w; w += G;
  int* counts2   = w; w += G;
  int* node_off2 = w; w += G;
  int* last_node = w; w += G;
  int* e_off     = w; w += G;
  int* is_last   = w; w += N;
  int* low       = w; w += N;
  int* rem       = w; w += N;
  int* emask     = w; w += E;
  int* epos      = w; w += E;
  int* br_mask   = w; w += N2;
  int* npos      = w; w += N2;
  int* remap     = w; w += N2;
  int* kmask     = w; w += E2;
  int* kpos      = w; w += E2;
  int* bsum      = w; w += 4096;
  int* counters  = w; w += 8;

  auto scan_ex = [&](const int* in, int* out, int n, int* total) {
    const int nb = cdiv_ll(n, SCAN_TILE);
    scan_block_k<<<nb, SCAN_T, 0, stream>>>(in, out, bsum, n);
    scan_top_k<<<1, SCAN_T, 0, stream>>>(bsum, nb, total);
    scan_add_k<<<nb, SCAN_T, 0, stream>>>(out, bsum, n);
  };

  // init
  fill_i32<<<cdiv_ll(G, TPB), TPB, 0, stream>>>(counts, 0, G);
  fill_i32<<<cdiv_ll(G, TPB), TPB, 0, stream>>>(counts2, 0, G);
  fill_i32<<<cdiv_ll(N, TPB), TPB, 0, stream>>>(is_last, 0, N);
  fill_i32<<<cdiv_ll(N, TPB), TPB, 0, stream>>>(rem, -1, N);
  fill_i32<<<cdiv_ll(E, TPB), TPB, 0, stream>>>(emask, 1, E);
  fill_i32<<<cdiv_ll(N2, TPB), TPB, 0, stream>>>(br_mask, 1, N2);

  // group bookkeeping
  hist_kernel<<<cdiv_ll(N, TPB), TPB, 0, stream>>>(batch, counts, N);
  hist_kernel<<<cdiv_ll(N2, TPB), TPB, 0, stream>>>(brb, counts2, N2);
  scan_ex(counts, node_off, G, counters + 3);
  scan_ex(counts2, node_off2, G, counters + 3);
  scan_ex(edge_num, e_off, G, counters + 3);
  lastnode_k<<<cdiv_ll(G, TPB), TPB, 0, stream>>>(counts, node_off, last_node, is_last, G, N);

  // scores via f32 WMMA, then gating + x_out
  score_wmma_k<<<cdiv_ll(N, 16), 32, 0, stream>>>(x, W, score, N, C);
  low_k<<<cdiv_ll(N, TPB), TPB, 0, stream>>>(score, is_last, low, N);
  if ((C & 3) == 0) {
    const long long n4 = (long long)baseNC / 4;
    xout_vec4_k<<<cdiv_ll(n4, TPB), TPB, 0, stream>>>((const float4*)x, score,
                                                      (float4*)dout, (int)n4, C / 4);
  } else {
    xout_s_k<<<cdiv_ll((long long)baseNC, TPB), TPB, 0, stream>>>(x, score, dout,
                                                                  (long long)baseNC, C);
  }

  // edge pruning + compaction
  cand_k<<<cdiv_ll(E, TPB), TPB, 0, stream>>>(ei, batch, e_off, last_node, low, rem, G, E);
  drop_k<<<cdiv_ll(N, TPB), TPB, 0, stream>>>(rem, emask, N);
  scan_ex(emask, epos, E, counters + 0);
  edge_idx_compact_k<<<cdiv_ll(E, TPB), TPB, 0, stream>>>(ei, emask, epos, counters,
                                                          dout, baseNC, E);
  inform_compact_k<<<cdiv_ll((long long)E * DE, TPB), TPB, 0, stream>>>(einf, emask, epos,
                                                                        counters, dout,
                                                                        baseNC, E, DE);
  write_enp_k<<<cdiv_ll(G, TPB), TPB, 0, stream>>>(e_off, edge_num, epos, counters,
                                                   dout, baseNC, G, E, DE);

  // branch-side node/edge compaction
  brdrop_k<<<cdiv_ll(N, TPB), TPB, 0, stream>>>(low, batch, node_off, node_off2, brc,
                                                br_mask, N, N2);
  scan_ex(br_mask, npos, N2, counters + 1);
  remap_k<<<cdiv_ll(N2, TPB), TPB, 0, stream>>>(br_mask, npos, remap, N2);
  brfeat_compact_k<<<cdiv_ll((long long)N2 * C, TPB), TPB, 0, stream>>>(brf, br_mask, npos,
                                                                        counters, dout, baseNC,
                                                                        G, DE, C, N2);
  keep_k<<<cdiv_ll(E2, TPB), TPB, 0, stream>>>(bri, remap, kmask, E2);
  scan_ex(kmask, kpos, E2, counters + 2);
  brindex_k<<<cdiv_ll(E2, TPB), TPB, 0, stream>>>(bri, remap, kmask, kpos, counters,
                                                  dout, baseNC, G, DE, C, E2);
  brbatch_k<<<cdiv_ll(N2, TPB), TPB, 0, stream>>>(brb, br_mask, npos, counters,
                                                  dout, baseNC, G, DE, C, N2);
}